// GCN_46523085750847
// MI455X (gfx1250) — compile-verified
//
#include <hip/hip_runtime.h>
#include <hip/hip_bf16.h>

typedef float v2f __attribute__((ext_vector_type(2)));
typedef float v8f __attribute__((ext_vector_type(8)));

#define F_IN  512
#define F_HID 32
#define F_OUT 64

// ---------------- degree / normalization ----------------

__global__ void k_init_deg(float* __restrict__ deg, int N) {
  int i = blockIdx.x * blockDim.x + threadIdx.x;
  if (i < N) deg[i] = 1.0f;  // self-loop contributes 1 to every node's degree
}

__global__ void k_count_deg(const int* __restrict__ dst, float* __restrict__ deg, int E) {
  int e = blockIdx.x * blockDim.x + threadIdx.x;
  if (e < E) unsafeAtomicAdd(&deg[dst[e]], 1.0f);
}

__global__ void k_rsqrt(float* __restrict__ deg, int N) {
  int i = blockIdx.x * blockDim.x + threadIdx.x;
  if (i < N) deg[i] = rsqrtf(deg[i]);  // deg >= 1 always (self loops)
}

// ---------------- GEMM1: H1[N,32] = X[N,512] @ W1[512,32] (f32 WMMA) ----------------

__global__ void k_gemm1(const float* __restrict__ X, const float* __restrict__ W,
                        float* __restrict__ H, int N) {
  const int lane = threadIdx.x & 31;
  const int wave = (blockIdx.x * blockDim.x + threadIdx.x) >> 5;
  const int m0 = wave << 4;
  if (m0 >= N) return;              // wave-uniform exit
  const int half = lane >> 4;       // 0: K={0,1}; 1: K={2,3} within each k-step
  const int lrow = lane & 15;
  int row = m0 + lrow; if (row >= N) row = N - 1;
  const float* xr = X + (size_t)row * F_IN;

  v8f c0 = {}; v8f c1 = {};
#pragma unroll 4
  for (int k0 = 0; k0 < F_IN; k0 += 4) {
    const int ka = k0 + (half << 1);
    v2f a;  a.x  = xr[ka];                      a.y  = xr[ka + 1];
    v2f b0; b0.x = W[ka * F_HID + lrow];        b0.y = W[(ka + 1) * F_HID + lrow];
    v2f b1; b1.x = W[ka * F_HID + 16 + lrow];   b1.y = W[(ka + 1) * F_HID + 16 + lrow];
    c0 = __builtin_amdgcn_wmma_f32_16x16x4_f32(false, a, false, b0, (short)0, c0, false, false);
    c1 = __builtin_amdgcn_wmma_f32_16x16x4_f32(false, a, false, b1, (short)0, c1, false, false);
  }
#pragma unroll
  for (int j = 0; j < 8; ++j) {
    int m = m0 + j + (half << 3);
    if (m < N) {
      float* hr = H + (size_t)m * F_HID;
      hr[lrow]      = c0[j];
      hr[16 + lrow] = c1[j];
    }
  }
}

// ---------------- GEMM2: H2[N,64] = relu(out1)[N,32] @ W2[32,64] (f32 WMMA) ----------------

__global__ void k_gemm2(const float* __restrict__ A, const float* __restrict__ W,
                        float* __restrict__ H, int N) {
  const int lane = threadIdx.x & 31;
  const int wave = (blockIdx.x * blockDim.x + threadIdx.x) >> 5;
  const int m0 = wave << 4;
  if (m0 >= N) return;
  const int half = lane >> 4;
  const int lrow = lane & 15;
  int row = m0 + lrow; if (row >= N) row = N - 1;
  const float* ar = A + (size_t)row * F_HID;

  v8f c0 = {}; v8f c1 = {}; v8f c2 = {}; v8f c3 = {};
#pragma unroll
  for (int k0 = 0; k0 < F_HID; k0 += 4) {
    const int ka = k0 + (half << 1);
    v2f a; a.x = fmaxf(ar[ka], 0.0f); a.y = fmaxf(ar[ka + 1], 0.0f);  // fused ReLU
    v2f b;
    b.x = W[ka * F_OUT + lrow];            b.y = W[(ka + 1) * F_OUT + lrow];
    c0 = __builtin_amdgcn_wmma_f32_16x16x4_f32(false, a, false, b, (short)0, c0, false, false);
    b.x = W[ka * F_OUT + 16 + lrow];       b.y = W[(ka + 1) * F_OUT + 16 + lrow];
    c1 = __builtin_amdgcn_wmma_f32_16x16x4_f32(false, a, false, b, (short)0, c1, false, false);
    b.x = W[ka * F_OUT + 32 + lrow];       b.y = W[(ka + 1) * F_OUT + 32 + lrow];
    c2 = __builtin_amdgcn_wmma_f32_16x16x4_f32(false, a, false, b, (short)0, c2, false, false);
    b.x = W[ka * F_OUT + 48 + lrow];       b.y = W[(ka + 1) * F_OUT + 48 + lrow];
    c3 = __builtin_amdgcn_wmma_f32_16x16x4_f32(false, a, false, b, (short)0, c3, false, false);
  }
#pragma unroll
  for (int j = 0; j < 8; ++j) {
    int m = m0 + j + (half << 3);
    if (m < N) {
      float* hr = H + (size_t)m * F_OUT;
      hr[lrow]      = c0[j];
      hr[16 + lrow] = c1[j];
      hr[32 + lrow] = c2[j];
      hr[48 + lrow] = c3[j];
    }
  }
}

// ---------------- aggregation (self-loop init + edge scatter) ----------------

__global__ void k_agg1_init(const float* __restrict__ h1, const float* __restrict__ dinv,
                            const float* __restrict__ bias, float* __restrict__ out1, int N) {
  int t = blockIdx.x * blockDim.x + threadIdx.x;
  if (t < N * F_HID) {
    int i = t >> 5, f = t & 31;
    float di = dinv[i];
    out1[t] = h1[t] * di * di + bias[f];   // self-loop term + b1 (commutes with edge adds)
  }
}

__global__ void k_agg1(const float* __restrict__ h1, const float* __restrict__ dinv,
                       const int* __restrict__ src, const int* __restrict__ dstv,
                       float* __restrict__ out1, int E) {
  int t = blockIdx.x * blockDim.x + threadIdx.x;
  int e = t >> 5;
  if (e >= E) return;
  int lane = t & 31;
  int s = src[e], d = dstv[e];
  float nrm = dinv[s] * dinv[d];
  float v = h1[(size_t)s * F_HID + lane] * nrm;
  unsafeAtomicAdd(&out1[(size_t)d * F_HID + lane], v);   // L2-resident hw f32 atomic
}

__global__ void k_agg2_init(const float* __restrict__ h2, const float* __restrict__ dinv,
                            const float* __restrict__ bias, float* __restrict__ out, int N) {
  int t = blockIdx.x * blockDim.x + threadIdx.x;
  if (t < N * F_OUT) {
    int i = t >> 6, f = t & 63;
    float di = dinv[i];
    out[t] = h2[t] * di * di + bias[f];
  }
}

__global__ void k_agg2(const float* __restrict__ h2, const float* __restrict__ dinv,
                       const int* __restrict__ src, const int* __restrict__ dstv,
                       float* __restrict__ out, int E) {
  int t = blockIdx.x * blockDim.x + threadIdx.x;
  int e = t >> 5;
  if (e >= E) return;
  int lane = t & 31;
  int s = src[e], d = dstv[e];
  float nrm = dinv[s] * dinv[d];
  const float* hs = h2 + (size_t)s * F_OUT;
  float* od = out + (size_t)d * F_OUT;
  unsafeAtomicAdd(&od[lane],      hs[lane]      * nrm);
  unsafeAtomicAdd(&od[lane + 32], hs[lane + 32] * nrm);
}

// ---------------- host launcher ----------------

static inline int cdiv_ll(long long a, long long b) { return (int)((a + b - 1) / b); }

extern "C" void kernel_launch(void* const* d_in, const int* in_sizes, int n_in,
                              void* d_out, int out_size, void* d_ws, size_t ws_size,
                              hipStream_t stream) {
  const float* X  = (const float*)d_in[0];
  const int*   ei = (const int*)d_in[1];
  const float* W1 = (const float*)d_in[2];
  const float* b1 = (const float*)d_in[3];
  const float* W2 = (const float*)d_in[4];
  const float* b2 = (const float*)d_in[5];
  float* out = (float*)d_out;

  const int N = in_sizes[0] / F_IN;   // 100000
  const int E = in_sizes[1] / 2;      // 3200000
  const int* src = ei;
  const int* dst = ei + E;

  // workspace layout: dinv[N] | h1[N*32] | out1[N*32] | h2[N*64]  (~52 MB)
  float* dinv = (float*)d_ws;
  float* h1   = dinv + N;
  float* out1 = h1 + (size_t)N * F_HID;
  float* h2   = out1 + (size_t)N * F_HID;

  const int B = 256;  // 8 wave32s per block
  const int mtiles = cdiv_ll(N, 16);

  k_init_deg <<<cdiv_ll(N, B), B, 0, stream>>>(dinv, N);
  k_count_deg<<<cdiv_ll(E, B), B, 0, stream>>>(dst, dinv, E);
  k_rsqrt    <<<cdiv_ll(N, B), B, 0, stream>>>(dinv, N);

  k_gemm1    <<<cdiv_ll((long long)mtiles * 32, B), B, 0, stream>>>(X, W1, h1, N);
  k_agg1_init<<<cdiv_ll((long long)N * F_HID, B), B, 0, stream>>>(h1, dinv, b1, out1, N);
  k_agg1     <<<cdiv_ll((long long)E * 32, B), B, 0, stream>>>(h1, dinv, src, dst, out1, E);

  k_gemm2    <<<cdiv_ll((long long)mtiles * 32, B), B, 0, stream>>>(out1, W2, h2, N);
  k_agg2_init<<<cdiv_ll((long long)N * F_OUT, B), B, 0, stream>>>(h2, dinv, b2, out, N);
  k_agg2     <<<cdiv_ll((long long)E * 32, B), B, 0, stream>>>(h2, dinv, src, dst, out, E);
}